// KalmanFilterCalculator_81810537055427
// MI455X (gfx1250) — compile-verified
//
#include <hip/hip_runtime.h>
#include <hip/hip_bf16.h>

// Batched Kalman filter + RTS smoother on gfx1250.
// One wave32 per batch element; 16x16 fp32 matmuls via chained
// V_WMMA_F32_16X16X4_F32 (native f32 tensor path on CDNA5).

#define DT_C 0.01f
constexpr int Bn = 2048;   // batch
constexpr int Tn = 128;    // time steps
constexpr int WAVES = 4;   // waves (batch elements) per block

typedef __attribute__((ext_vector_type(2))) float v2f;
typedef __attribute__((ext_vector_type(8))) float v8f;

__device__ __forceinline__ void wsync() { __builtin_amdgcn_wave_barrier(); }

__device__ __forceinline__ v8f wmma4(v2f a, v2f b, v8f c) {
  // D = A(16x4) x B(4x16) + C(16x16), fp32. 8-arg form:
  // (neg_a, A, neg_b, B, c_mod, C, reuse_a, reuse_b); f32 WMMA: neg bits must be 0.
  return __builtin_amdgcn_wmma_f32_16x16x4_f32(false, a, false, b, (short)0, c,
                                               false, false);
}

// C += op(A) * op(B), 16x16x16, from row-major LDS with leading dims lda/ldb.
// A-layout (16x4 f32): lane<16 -> row M=lane, K={k0,k0+1}; lane>=16 -> K={k0+2,k0+3}.
// B-layout (4x16 f32): VGPR0 holds K=k0 (lanes 0-15) / K=k0+2 (lanes 16-31), etc.
template <bool TA, bool TB>
__device__ __forceinline__ v8f mm16(const float* A, int lda, const float* B,
                                    int ldb, v8f acc) {
  const int lane = threadIdx.x & 31;
  const int half = lane >> 4;
  const int m = lane & 15;
#pragma unroll
  for (int k0 = 0; k0 < 16; k0 += 4) {
    const int ka = k0 + half * 2;
    v2f a, b;
    if (TA) { a.x = A[(ka + 0) * lda + m]; a.y = A[(ka + 1) * lda + m]; }
    else    { a.x = A[m * lda + ka + 0];   a.y = A[m * lda + ka + 1];   }
    if (TB) { b.x = B[m * ldb + ka + 0];   b.y = B[m * ldb + ka + 1];   }
    else    { b.x = B[(ka + 0) * ldb + m]; b.y = B[(ka + 1) * ldb + m]; }
    acc = wmma4(a, b, acc);
  }
  return acc;
}

// C/D matrix layout: VGPR r -> lanes 0-15: M=r, N=lane; lanes 16-31: M=r+8.
__device__ __forceinline__ v8f load_c(const float* M, int ld) {
  const int lane = threadIdx.x & 31;
  const int half = lane >> 4, nn = lane & 15;
  v8f c;
#pragma unroll
  for (int r = 0; r < 8; ++r) c[r] = M[(r + 8 * half) * ld + nn];
  return c;
}

template <bool NEG>
__device__ __forceinline__ void store_c(float* M, int ld, v8f c) {
  const int lane = threadIdx.x & 31;
  const int half = lane >> 4, nn = lane & 15;
#pragma unroll
  for (int r = 0; r < 8; ++r) M[(r + 8 * half) * ld + nn] = NEG ? -c[r] : c[r];
}

// Gauss-Jordan inverse of a 16x16 SPD matrix (no pivoting needed).
// aug: 16x32 LDS scratch; result S^-1 lands in the right half (stride 32).
__device__ void gj_inv16(float* aug, const float* Min) {
  const int lane = threadIdx.x & 31;
  for (int r = 0; r < 16; ++r) {
    float v = (lane < 16) ? Min[r * 16 + lane] : ((lane - 16) == r ? 1.f : 0.f);
    aug[r * 32 + lane] = v;
  }
  wsync();
  for (int k = 0; k < 16; ++k) {
    float invp = 1.0f / aug[k * 32 + k];
    float pk = aug[k * 32 + lane] * invp;
    aug[k * 32 + lane] = pk;
    wsync();
    for (int r = 0; r < 16; ++r) {
      if (r != k) {
        float f = aug[r * 32 + k];
        aug[r * 32 + lane] -= f * pk;
      }
    }
    wsync();
  }
}

__global__ void __launch_bounds__(WAVES * 32)
kalman_rts_kernel(const float* __restrict__ state0, const float* __restrict__ P0,
                  const float* __restrict__ controls, const float* __restrict__ obs,
                  const float* __restrict__ Ain, const float* __restrict__ Bcin,
                  const float* __restrict__ Hin, const float* __restrict__ Qin,
                  const float* __restrict__ Rin, float* __restrict__ out,
                  float* __restrict__ ws) {
  __shared__ float sF[256];    // F = I + DT*A
  __shared__ float sHp[256];   // H zero-padded to 16x16 (rows 8..15 = 0)
  __shared__ float sQ[256];
  __shared__ float sRb[256];   // blockdiag(R, I)  -> S_pad = blockdiag(S, I)
  __shared__ float sBcDT[64];  // DT * Bc (16x4)
  __shared__ float wbuf[WAVES][2112];

  const int tid = threadIdx.x;
  for (int i = tid; i < 256; i += blockDim.x) {
    int r = i >> 4, c = i & 15;
    sF[i] = ((r == c) ? 1.f : 0.f) + DT_C * Ain[i];
    sHp[i] = (r < 8) ? Hin[r * 16 + c] : 0.f;
    sQ[i] = Qin[i];
    sRb[i] = (r < 8 && c < 8) ? Rin[r * 8 + c] : ((r == c) ? 1.f : 0.f);
  }
  for (int i = tid; i < 64; i += blockDim.x) sBcDT[i] = DT_C * Bcin[i];
  __syncthreads();

  const int w = tid >> 5, lane = tid & 31;
  const int b = blockIdx.x * WAVES + w;

  float* Pf  = &wbuf[w][0];     // filtered covariance
  float* Pp  = &wbuf[w][256];   // predicted covariance
  float* T1  = &wbuf[w][512];
  float* T2  = &wbuf[w][768];
  float* Kn  = &wbuf[w][1024];  // forward: -K ; backward: G
  float* Ps  = &wbuf[w][1280];  // smoothed covariance
  float* Aug = &wbuf[w][1536];  // 16x32 Gauss-Jordan scratch
  float* vec = &wbuf[w][2048];  // sf | sp/ss | iv | dv
  float* sf = vec;
  float* sp = vec + 16;
  float* iv = vec + 32;
  float* dv = vec + 48;

  if (lane < 16) sf[lane] = state0[(size_t)b * 16 + lane];
  { v8f c0 = load_c(&P0[(size_t)b * 256], 16); store_c<false>(Pf, 16, c0); }
  wsync();

  // Workspace: s_f sequence (16 MB) + P_f sequence (268 MB).
  // P_p[t+1], s_p[t+1] are recomputed in the backward pass from P_f[t], s_f[t].
  float* sfs = ws;                            // Tn*Bn*16 floats
  float* pfs = ws + (size_t)Tn * Bn * 16;     // Tn*Bn*256 floats

  // ----------------- forward Kalman filter -----------------
  for (int t = 0; t < Tn; ++t) {
    const float* u = &controls[((size_t)b * Tn + t) * 4];
    const float* y = &obs[((size_t)b * Tn + t) * 8];

    // s_p = F s_f + DT*Bc u
    if (lane < 16) {
      float acc = 0.f;
      for (int j = 0; j < 16; ++j) acc += sF[lane * 16 + j] * sf[j];
      for (int k = 0; k < 4; ++k) acc += sBcDT[lane * 4 + k] * u[k];
      sp[lane] = acc;
    }
    wsync();

    // Pp = F Pf F^T + Q
    { v8f c = {}; c = mm16<false, false>(sF, 16, Pf, 16, c); store_c<false>(T1, 16, c); }
    wsync();
    { v8f c = load_c(sQ, 16); c = mm16<false, true>(T1, 16, sF, 16, c); store_c<false>(Pp, 16, c); }
    wsync();

    // PHt = Pp Hp^T   (cols 8..15 are zero)
    { v8f c = {}; c = mm16<false, true>(Pp, 16, sHp, 16, c); store_c<false>(T1, 16, c); }
    wsync();

    // S_pad = Hp PHt + blockdiag(R, I)
    { v8f c = load_c(sRb, 16); c = mm16<false, false>(sHp, 16, T1, 16, c); store_c<false>(T2, 16, c); }
    wsync();
    gj_inv16(Aug, T2);  // Aug right half = blockdiag(S^-1, I)
    wsync();

    // Kn = -(PHt * Sinv)   (f32 WMMA has no A-negate -> keep -K)
    { v8f c = {}; c = mm16<false, false>(T1, 16, Aug + 16, 32, c); store_c<true>(Kn, 16, c); }
    wsync();

    // innovation = y - H s_p (padded with zeros)
    if (lane < 16) {
      float val = 0.f;
      if (lane < 8) {
        float acc = y[lane];
        for (int j = 0; j < 16; ++j) acc -= sHp[lane * 16 + j] * sp[j];
        val = acc;
      }
      iv[lane] = val;
    }
    wsync();

    // s_f = s_p + K innov = s_p - Kn innov
    if (lane < 16) {
      float acc = sp[lane];
      for (int j = 0; j < 8; ++j) acc -= Kn[lane * 16 + j] * iv[j];
      sf[lane] = acc;
    }

    // T2 = Kn*Hp = -(K H)
    { v8f c = {}; c = mm16<false, false>(Kn, 16, sHp, 16, c); store_c<false>(T2, 16, c); }
    wsync();

    // Pf = Pp + T2*Pp = Pp - K H Pp ; spill to HBM for smoother
    {
      v8f c = load_c(Pp, 16);
      c = mm16<false, false>(T2, 16, Pp, 16, c);
      store_c<false>(Pf, 16, c);
      const int half = lane >> 4, nn = lane & 15;
      float* dst = &pfs[((size_t)t * Bn + b) * 256];
#pragma unroll
      for (int r = 0; r < 8; ++r) dst[(r + 8 * half) * 16 + nn] = c[r];
    }
    wsync();
    if (lane < 16) sfs[((size_t)t * Bn + b) * 16 + lane] = sf[lane];
    wsync();
  }

  // ----------------- backward RTS smoother -----------------
  { v8f c = load_c(Pf, 16); store_c<false>(Ps, 16, c); }
  float* ss = sp;  // reuse slot: carries s_s_next
  if (lane < 16) {
    ss[lane] = sf[lane];
    out[((size_t)b * Tn + (Tn - 1)) * 16 + lane] = sf[lane];
  }
  wsync();

  for (int t = Tn - 2; t >= 0; --t) {
    if (lane < 16) sf[lane] = sfs[((size_t)t * Bn + b) * 16 + lane];
    { v8f c = load_c(&pfs[((size_t)t * Bn + b) * 256], 16); store_c<false>(Pf, 16, c); }
    wsync();

    // dv = s_s_next - s_p[t+1],  s_p[t+1] = F s_f[t] + DT*Bc u[t+1]
    const float* u = &controls[((size_t)b * Tn + (t + 1)) * 4];
    if (lane < 16) {
      float acc = 0.f;
      for (int j = 0; j < 16; ++j) acc += sF[lane * 16 + j] * sf[j];
      for (int k = 0; k < 4; ++k) acc += sBcDT[lane * 4 + k] * u[k];
      dv[lane] = ss[lane] - acc;
    }

    // Pp[t+1] = F Pf[t] F^T + Q   (recompute instead of storing)
    { v8f c = {}; c = mm16<false, false>(sF, 16, Pf, 16, c); store_c<false>(T1, 16, c); }
    wsync();
    { v8f c = load_c(sQ, 16); c = mm16<false, true>(T1, 16, sF, 16, c); store_c<false>(Pp, 16, c); }
    wsync();
    gj_inv16(Aug, Pp);  // Pinv
    wsync();

    // G = Pf F^T Pinv  -> Kn
    { v8f c = {}; c = mm16<false, true>(Pf, 16, sF, 16, c); store_c<false>(T1, 16, c); }
    wsync();
    { v8f c = {}; c = mm16<false, false>(T1, 16, Aug + 16, 32, c); store_c<false>(Kn, 16, c); }
    wsync();

    // D = Ps_next - Pp
    for (int i2 = lane; i2 < 256; i2 += 32) T2[i2] = Ps[i2] - Pp[i2];
    wsync();

    // s_s = sf + G dv
    if (lane < 16) {
      float acc = sf[lane];
      for (int j = 0; j < 16; ++j) acc += Kn[lane * 16 + j] * dv[j];
      ss[lane] = acc;
      out[((size_t)b * Tn + t) * 16 + lane] = acc;
    }

    // Ps = Pf + (G D) G^T
    { v8f c = {}; c = mm16<false, false>(Kn, 16, T2, 16, c); store_c<false>(T1, 16, c); }
    wsync();
    { v8f c = load_c(Pf, 16); c = mm16<false, true>(T1, 16, Kn, 16, c); store_c<false>(Ps, 16, c); }
    wsync();
  }
}

extern "C" void kernel_launch(void* const* d_in, const int* in_sizes, int n_in,
                              void* d_out, int out_size, void* d_ws, size_t ws_size,
                              hipStream_t stream) {
  (void)in_sizes; (void)n_in; (void)out_size; (void)ws_size;
  const float* state0   = (const float*)d_in[0];
  const float* P0       = (const float*)d_in[1];
  const float* controls = (const float*)d_in[2];
  const float* obs      = (const float*)d_in[3];
  const float* A        = (const float*)d_in[4];
  const float* Bc       = (const float*)d_in[5];
  const float* H        = (const float*)d_in[6];
  const float* Q        = (const float*)d_in[7];
  const float* R        = (const float*)d_in[8];
  float* out = (float*)d_out;
  float* ws  = (float*)d_ws;

  dim3 grid(Bn / WAVES);
  dim3 block(WAVES * 32);
  hipLaunchKernelGGL(kalman_rts_kernel, grid, block, 0, stream, state0, P0,
                     controls, obs, A, Bc, H, Q, R, out, ws);
}